// GraphConvLayer_61083024884222
// MI455X (gfx1250) — compile-verified
//
#include <hip/hip_runtime.h>

#define NN 100000
#define EE 1600000
#define DD 64

typedef __attribute__((ext_vector_type(2))) float v2f;
typedef __attribute__((ext_vector_type(8))) float v8f;

// ---------------------------------------------------------------------------
// Kernel 1: zero the aggregation buffer (in workspace)
// ---------------------------------------------------------------------------
__global__ __launch_bounds__(256) void gc_zero_kernel(float4* __restrict__ agg4) {
    unsigned i = blockIdx.x * 256u + threadIdx.x;          // NN*DD/4 threads exactly
    agg4[i] = float4{0.f, 0.f, 0.f, 0.f};
}

// ---------------------------------------------------------------------------
// Kernel 2: edge scatter  agg[dst] += x[src] * edge_attr
// 16 lanes per edge, each lane handles 4 contiguous floats (float4 gather).
// Atomics forced to hardware no-return global_atomic_add_f32 (device scope,
// executed at the L2 atomic units) via inline asm so the compiler cannot
// expand them into a CAS retry loop.
// ---------------------------------------------------------------------------
__device__ __forceinline__ void atomic_fadd_hw(float* p, float v) {
    asm volatile("global_atomic_add_f32 %0, %1, off scope:SCOPE_DEV"
                 :: "v"(p), "v"(v) : "memory");
}

__global__ __launch_bounds__(256) void gc_scatter_kernel(
    const float* __restrict__ x,
    const long long* __restrict__ edge_index,   // [2, E] int64, row0=src row1=dst
    const float* __restrict__ edge_attr,
    float* __restrict__ agg) {
    long long t = (long long)blockIdx.x * 256 + threadIdx.x;
    long long e = t >> 4;
    int p = (int)(t & 15);
    if (e >= (long long)EE) return;

    long long s = edge_index[e];
    long long d = edge_index[(long long)EE + e];
    float w = edge_attr[e];

    float4 v = *(const float4*)(x + s * DD + p * 4);
    float* dp = agg + d * DD + p * 4;
    atomic_fadd_hw(dp + 0, v.x * w);
    atomic_fadd_hw(dp + 1, v.y * w);
    atomic_fadd_hw(dp + 2, v.z * w);
    atomic_fadd_hw(dp + 3, v.w * w);
}

// ---------------------------------------------------------------------------
// Kernel 3: fused  out = relu(agg @ W_rel^T + b_rel + x @ W_root^T)
// via V_WMMA_F32_16X16X4_F32. One wave -> 16 rows x 64 cols.
// Weights staged in LDS (row-major [n][k], 64 floats per row).
// ---------------------------------------------------------------------------
__global__ __launch_bounds__(256) void gc_gemm_kernel(
    const float* __restrict__ x,
    const float* __restrict__ agg,
    const float* __restrict__ W_rel,
    const float* __restrict__ b_rel,
    const float* __restrict__ W_root,
    float* __restrict__ out) {
    __shared__ float sW_rel[DD * DD];
    __shared__ float sW_root[DD * DD];

    const int tid = threadIdx.x;
    // cooperative load of both 64x64 weight matrices (float4 per thread x4 iters)
    for (int i = tid * 4; i < DD * DD; i += 256 * 4) {
        *(float4*)(sW_rel + i)  = *(const float4*)(W_rel + i);
        *(float4*)(sW_root + i) = *(const float4*)(W_root + i);
    }
    __syncthreads();

    const int wave = tid >> 5;              // 0..7
    const int lane = tid & 31;
    const int m0 = (blockIdx.x * 8 + wave) * 16;   // 16-row tile base
    if (m0 >= NN) return;                   // wave-uniform: EXEC stays all-ones

    const int lr = lane & 15;               // M / N index within tile
    const int lg = lane >> 4;               // half-wave group (K phase)

    v8f acc[4] = {v8f{}, v8f{}, v8f{}, v8f{}};

    const float* aggp = agg + (size_t)(m0 + lr) * DD + 2 * lg;
    const float* xp   = x   + (size_t)(m0 + lr) * DD + 2 * lg;

#pragma unroll
    for (int kk = 0; kk < 16; ++kk) {       // K = 64 in steps of 4
        v2f a_agg = *(const v2f*)(aggp + kk * 4);
        v2f a_x   = *(const v2f*)(xp   + kk * 4);
#pragma unroll
        for (int nt = 0; nt < 4; ++nt) {    // four 16-col tiles
            const int n = nt * 16 + lr;
            v2f b_rel2  = *(const v2f*)(sW_rel  + n * DD + kk * 4 + 2 * lg);
            v2f b_root2 = *(const v2f*)(sW_root + n * DD + kk * 4 + 2 * lg);
            acc[nt] = __builtin_amdgcn_wmma_f32_16x16x4_f32(
                false, a_agg, false, b_rel2, (short)0, acc[nt], false, false);
            acc[nt] = __builtin_amdgcn_wmma_f32_16x16x4_f32(
                false, a_x,   false, b_root2, (short)0, acc[nt], false, false);
        }
    }

    // epilogue: bias + relu + store.
    // C/D layout: VGPR r -> row m0 + r + 8*lg, col n = nt*16 + lr
#pragma unroll
    for (int nt = 0; nt < 4; ++nt) {
        const int n = nt * 16 + lr;
        const float bias = b_rel[n];
#pragma unroll
        for (int r = 0; r < 8; ++r) {
            const int m = m0 + r + 8 * lg;
            float v = acc[nt][r] + bias;
            out[(size_t)m * DD + n] = v > 0.f ? v : 0.f;
        }
    }
}

// ---------------------------------------------------------------------------
extern "C" void kernel_launch(void* const* d_in, const int* in_sizes, int n_in,
                              void* d_out, int out_size, void* d_ws, size_t ws_size,
                              hipStream_t stream) {
    const float* x          = (const float*)d_in[0];
    const long long* eindex = (const long long*)d_in[1];
    const float* eattr      = (const float*)d_in[2];
    const float* W_rel      = (const float*)d_in[3];
    const float* b_rel      = (const float*)d_in[4];
    const float* W_root     = (const float*)d_in[5];
    float* out              = (float*)d_out;
    float* agg              = (float*)d_ws;   // NN*DD floats = 25.6 MB scratch

    // 1) zero agg
    gc_zero_kernel<<<(NN * DD / 4) / 256, 256, 0, stream>>>((float4*)agg);

    // 2) scatter edges (16 lanes per edge, hardware f32 atomics)
    long long sthreads = (long long)EE * 16;
    gc_scatter_kernel<<<(unsigned)((sthreads + 255) / 256), 256, 0, stream>>>(
        x, eindex, eattr, agg);

    // 3) fused dual-GEMM + bias + relu  (6250 row-tiles, 8 waves/block)
    gc_gemm_kernel<<<(6250 + 7) / 8, 256, 0, stream>>>(
        x, agg, W_rel, b_rel, W_root, out);
}